// AddComp_89378269430082
// MI455X (gfx1250) — compile-verified
//
#include <hip/hip_runtime.h>
#include <math.h>

// out[b,m,n] = sum_h tanh(q[b,m,h] + k[b,n,h]) * v[h]
// B=4, M=N=1024, H=128.  Compute-bound on transcendental tanh; WMMA is
// mathematically inapplicable (elementwise tanh between rank-1 structure and
// the h-contraction).  Use gfx1250 v_tanh_f32 + async global->LDS staging.

#define B_SZ 4
#define M_DIM 1024
#define N_DIM 1024
#define H_DIM 128

#define BM 64            // m-tile per workgroup
#define BN 64            // n-tile per workgroup
#define TS 132           // padded LDS row stride in floats (bank-conflict free)

#if defined(__has_builtin)
#if __has_builtin(__builtin_amdgcn_tanhf)
#define TANH_F32(x) __builtin_amdgcn_tanhf(x)
#endif
#endif
#ifndef TANH_F32
#define TANH_F32(x) tanhf(x)
#endif

// CDNA5 async copy: GLOBAL_LOAD_ASYNC_TO_LDS_B128 (tracked by ASYNCcnt).
// vdst = VGPR holding LDS byte address, vaddr = 64-bit global address.
__device__ __forceinline__ void async_copy16(unsigned lds_byte_addr,
                                             const float* gsrc) {
  asm volatile("global_load_async_to_lds_b128 %0, %1, off"
               :: "v"(lds_byte_addr), "v"(gsrc)
               : "memory");
}

__device__ __forceinline__ void wait_async_all() {
  asm volatile("s_wait_asynccnt 0" ::: "memory");
}

__global__ __launch_bounds__(256) void
tanh_outer_reduce_kernel(const float* __restrict__ q,
                         const float* __restrict__ k,
                         const float* __restrict__ v,
                         float* __restrict__ out) {
  __shared__ __align__(16) float sQ[BM * TS];        // [m][h], padded rows
  __shared__ __align__(16) float sK[BN * TS];        // [n][h], padded rows
  __shared__ __align__(16) float sV[H_DIM];

  const int t  = threadIdx.x;
  const int n0 = blockIdx.x * BN;
  const int m0 = blockIdx.y * BM;
  const int b  = blockIdx.z;

  const float* qb = q + ((size_t)b * M_DIM + m0) * H_DIM;
  const float* kb = k + ((size_t)b * N_DIM + n0) * H_DIM;

  // ---- Stage tiles into LDS via CDNA5 async loads --------------------------
  // Each tile is 64 rows x 128 floats = 2048 16-byte chunks; 8 per thread.
  // Global side is fully contiguous (chunk c -> bytes [16c, 16c+16)).
  // LDS side inserts +4 floats of padding per row (stride TS=132).
#pragma unroll
  for (int it = 0; it < 8; ++it) {
    const int c  = t + it * 256;       // chunk id 0..2047
    const int r  = c >> 5;             // row (32 chunks per 128-float row)
    const int c4 = c & 31;             // 16B chunk within row
    const unsigned qdst = (unsigned)(uintptr_t)&sQ[r * TS + c4 * 4];
    const unsigned kdst = (unsigned)(uintptr_t)&sK[r * TS + c4 * 4];
    async_copy16(qdst, qb + 4 * c);
    async_copy16(kdst, kb + 4 * c);
  }
  if (t < 32) {
    const unsigned vdst = (unsigned)(uintptr_t)&sV[t * 4];
    async_copy16(vdst, v + 4 * t);
  }
  wait_async_all();
  __syncthreads();

  // ---- Compute 4x4 micro-tile per thread -----------------------------------
  // thread (tx,ty): m = m0 + 4*ty + i (i<4), n = n0 + tx + 16*j (j<4)
  const int tx = t & 15;
  const int ty = t >> 4;

  float acc[4][4] = {};

  for (int h = 0; h < H_DIM; h += 4) {
    float4 qv[4], kv[4];
#pragma unroll
    for (int i = 0; i < 4; ++i)
      qv[i] = *(const float4*)&sQ[(4 * ty + i) * TS + h];   // broadcast reads
#pragma unroll
    for (int j = 0; j < 4; ++j)
      kv[j] = *(const float4*)&sK[(tx + 16 * j) * TS + h];  // conflict-free
    const float4 vv = *(const float4*)&sV[h];

#pragma unroll
    for (int i = 0; i < 4; ++i) {
#pragma unroll
      for (int j = 0; j < 4; ++j) {
        float a = acc[i][j];
        a = fmaf(vv.x, TANH_F32(qv[i].x + kv[j].x), a);
        a = fmaf(vv.y, TANH_F32(qv[i].y + kv[j].y), a);
        a = fmaf(vv.z, TANH_F32(qv[i].z + kv[j].z), a);
        a = fmaf(vv.w, TANH_F32(qv[i].w + kv[j].w), a);
        acc[i][j] = a;
      }
    }
  }

  // ---- Store (coalesced: lanes sweep consecutive n) ------------------------
  float* ob = out + ((size_t)b * M_DIM + m0) * (size_t)N_DIM + n0;
#pragma unroll
  for (int i = 0; i < 4; ++i) {
#pragma unroll
    for (int j = 0; j < 4; ++j) {
      ob[(size_t)(4 * ty + i) * N_DIM + tx + 16 * j] = acc[i][j];
    }
  }
}

extern "C" void kernel_launch(void* const* d_in, const int* in_sizes, int n_in,
                              void* d_out, int out_size, void* d_ws, size_t ws_size,
                              hipStream_t stream) {
  (void)in_sizes; (void)n_in; (void)out_size; (void)d_ws; (void)ws_size;
  const float* q = (const float*)d_in[0];  // (B, M, H)
  const float* k = (const float*)d_in[1];  // (B, N, H)
  const float* v = (const float*)d_in[2];  // (H,)
  float* out = (float*)d_out;              // (B, M, N)

  dim3 grid(N_DIM / BN, M_DIM / BM, B_SZ);
  dim3 block(256);
  tanh_outer_reduce_kernel<<<grid, block, 0, stream>>>(q, k, v, out);
}